// CommunityDetection_7421703488232
// MI455X (gfx1250) — compile-verified
//
#include <hip/hip_runtime.h>

typedef float v2f __attribute__((ext_vector_type(2)));
typedef float v8f __attribute__((ext_vector_type(8)));

#define N_NODES 50000
#define N_EDGES 1600000
#define IN_F    128
#define HID_F   64
#define OUT_F   16

// ---------------------------------------------------------------- utilities

__global__ void gc_zero_kernel(float* __restrict__ p, long n) {
    long i = (long)blockIdx.x * blockDim.x + threadIdx.x;
    long stride = (long)gridDim.x * blockDim.x;
    for (; i < n; i += stride) p[i] = 0.0f;
}

__global__ void gc_degree_kernel(const int* __restrict__ src,
                                 const int* __restrict__ dst,
                                 float* __restrict__ deg_out,
                                 float* __restrict__ deg_in) {
    int i = blockIdx.x * blockDim.x + threadIdx.x;
    if (i < N_EDGES) {
        atomicAdd(&deg_out[src[i]], 1.0f);
        atomicAdd(&deg_in[dst[i]], 1.0f);
    }
}

__global__ void gc_norm_kernel(float* __restrict__ deg_out,
                               float* __restrict__ deg_in) {
    int i = blockIdx.x * blockDim.x + threadIdx.x;
    if (i < N_NODES) {
        deg_out[i] = rsqrtf(fmaxf(deg_out[i], 1.0f));   // norm_s
        deg_in[i]  = rsqrtf(fmaxf(deg_in[i], 1.0f));    // norm_d
    }
}

// -------------------------------------------------- GEMM1: (x*norm_s) @ W1
// [50000,128] @ [128,64] -> h1 [50000,64], fp32 WMMA 16x16x4.
// Block = 128 threads (4 waves). Each block: one 16-row tile; wave w does
// output columns [16w, 16w+16). grid = 50000/16 = 3125.
__global__ void gc_gemm1_wmma(const float* __restrict__ x,
                              const float* __restrict__ W1,
                              const float* __restrict__ norm_s,
                              float* __restrict__ h1) {
    const int wave  = threadIdx.x >> 5;       // 0..3 -> column tile
    const int lane  = threadIdx.x & 31;
    const int lhalf = lane & 15;
    const int khi   = (lane >> 4) << 1;       // 0 for lanes 0-15, 2 for 16-31
    const int row0  = blockIdx.x * 16;
    const int n0    = wave * 16;

    // A-matrix 16x4 f32 layout: lane L holds row M=L%16;
    // VGPR0 = K=(khi+0), VGPR1 = K=(khi+1)
    const int   arow  = row0 + lhalf;
    const float ns    = norm_s[arow];
    const float* ap   = x + (long)arow * IN_F + khi;

    v8f c = {};
    #pragma unroll 4
    for (int k0 = 0; k0 < IN_F; k0 += 4) {
        v2f a, b;
        a.x = ap[k0 + 0] * ns;
        a.y = ap[k0 + 1] * ns;
        // B-matrix 4x16 f32: lane L holds col N=L%16; K split like A
        b.x = W1[(long)(k0 + khi + 0) * HID_F + n0 + lhalf];
        b.y = W1[(long)(k0 + khi + 1) * HID_F + n0 + lhalf];
        c = __builtin_amdgcn_wmma_f32_16x16x4_f32(false, a, false, b,
                                                  (short)0, c, false, false);
    }

    // C/D layout: VGPR r -> M = r (lanes 0-15) / r+8 (lanes 16-31), N = lane%16
    const int mbase = row0 + ((lane >> 4) << 3);
    const int col   = n0 + lhalf;
    #pragma unroll
    for (int r = 0; r < 8; ++r)
        h1[(long)(mbase + r) * HID_F + col] = c[r];
}

// -------------------------------------------------- edge scatter-aggregate
// agg[dst[e]][f] += h[src[e]][f]; one thread per (edge, 4-float group).
__global__ void gc_agg_kernel(const int* __restrict__ src,
                              const int* __restrict__ dst,
                              const float* __restrict__ h,
                              float* __restrict__ agg,
                              int feat) {
    const int  groups = feat >> 2;                 // float4 groups per row
    const long total  = (long)N_EDGES * groups;
    long tid    = (long)blockIdx.x * blockDim.x + threadIdx.x;
    long stride = (long)gridDim.x * blockDim.x;
    for (; tid < total; tid += stride) {
        const int e = (int)(tid / groups);
        const int g = (int)(tid % groups) << 2;
        const float4 v = *(const float4*)(h + (long)src[e] * feat + g);
        float* o = agg + (long)dst[e] * feat + g;
        atomicAdd(o + 0, v.x);
        atomicAdd(o + 1, v.y);
        atomicAdd(o + 2, v.z);
        atomicAdd(o + 3, v.w);
    }
}

// -------------------------------------------------- GEMM2 (fused act):
// A = relu(agg1*norm_d + b1)  [50000,64] @ W2 [64,16] -> h2 [50000,16]
// One wave per 16-row tile (N=16=OUT exactly). Block=256 (8 waves).
__global__ void gc_gemm2_wmma(const float* __restrict__ agg1,
                              const float* __restrict__ W2,
                              const float* __restrict__ norm_d,
                              const float* __restrict__ b1,
                              float* __restrict__ h2) {
    const int wtile = blockIdx.x * (blockDim.x >> 5) + (threadIdx.x >> 5);
    if (wtile >= N_NODES / 16) return;             // wave-uniform exit
    const int lane  = threadIdx.x & 31;
    const int lhalf = lane & 15;
    const int khi   = (lane >> 4) << 1;
    const int row0  = wtile * 16;

    const int   arow = row0 + lhalf;
    const float nd   = norm_d[arow];
    const float* ap  = agg1 + (long)arow * HID_F + khi;

    v8f c = {};
    #pragma unroll
    for (int k0 = 0; k0 < HID_F; k0 += 4) {
        v2f a, b;
        a.x = fmaxf(ap[k0 + 0] * nd + b1[k0 + khi + 0], 0.0f);
        a.y = fmaxf(ap[k0 + 1] * nd + b1[k0 + khi + 1], 0.0f);
        b.x = W2[(long)(k0 + khi + 0) * OUT_F + lhalf];
        b.y = W2[(long)(k0 + khi + 1) * OUT_F + lhalf];
        c = __builtin_amdgcn_wmma_f32_16x16x4_f32(false, a, false, b,
                                                  (short)0, c, false, false);
    }

    const int mbase = row0 + ((lane >> 4) << 3);
    #pragma unroll
    for (int r = 0; r < 8; ++r)
        h2[(long)(mbase + r) * OUT_F + lhalf] = c[r];
}

// -------------------------------------------------- finalize layer 2
__global__ void gc_finalize_kernel(const float* __restrict__ agg2,
                                   const float* __restrict__ norm_d,
                                   const float* __restrict__ b2,
                                   float* __restrict__ out) {
    long i = (long)blockIdx.x * blockDim.x + threadIdx.x;
    if (i < (long)N_NODES * OUT_F) {
        const int n = (int)(i / OUT_F);
        const int f = (int)(i % OUT_F);
        out[i] = agg2[i] * norm_d[n] + b2[f];
    }
}

// ---------------------------------------------------------------- launcher

extern "C" void kernel_launch(void* const* d_in, const int* in_sizes, int n_in,
                              void* d_out, int out_size, void* d_ws, size_t ws_size,
                              hipStream_t stream) {
    const float* features = (const float*)d_in[0];   // [N, 128]
    const float* W1       = (const float*)d_in[1];   // [128, 64]
    const float* b1       = (const float*)d_in[2];   // [64]
    const float* W2       = (const float*)d_in[3];   // [64, 16]
    const float* b2       = (const float*)d_in[4];   // [16]
    const int*   src      = (const int*)d_in[5];     // [E]
    const int*   dst      = (const int*)d_in[6];     // [E]
    float*       out      = (float*)d_out;           // [N, 16]

    // workspace layout (floats)
    float* ws     = (float*)d_ws;
    float* norm_s = ws;                               // N    (deg_out -> norm_s)
    float* norm_d = norm_s + N_NODES;                 // N    (deg_in  -> norm_d)
    float* h1     = norm_d + N_NODES;                 // N*64
    float* agg1   = h1 + (long)N_NODES * HID_F;       // N*64
    float* h2     = agg1 + (long)N_NODES * HID_F;     // N*16
    float* agg2   = h2 + (long)N_NODES * OUT_F;       // N*16

    const int T = 256;

    // zero degree counters + aggregation buffers (deterministic every call)
    gc_zero_kernel<<<512, T, 0, stream>>>(norm_s, 2L * N_NODES);
    gc_zero_kernel<<<2048, T, 0, stream>>>(agg1, (long)N_NODES * HID_F);
    gc_zero_kernel<<<1024, T, 0, stream>>>(agg2, (long)N_NODES * OUT_F);

    gc_degree_kernel<<<(N_EDGES + T - 1) / T, T, 0, stream>>>(src, dst, norm_s, norm_d);
    gc_norm_kernel<<<(N_NODES + T - 1) / T, T, 0, stream>>>(norm_s, norm_d);

    // layer 1: project -> aggregate
    gc_gemm1_wmma<<<N_NODES / 16, 128, 0, stream>>>(features, W1, norm_s, h1);
    {
        long total = (long)N_EDGES * (HID_F / 4);
        int blocks = (int)((total + T - 1) / T);
        gc_agg_kernel<<<blocks, T, 0, stream>>>(src, dst, h1, agg1, HID_F);
    }

    // layer 2: finalize+relu fused into projection -> aggregate -> finalize
    {
        int waves_per_block = T / 32;
        int blocks = (N_NODES / 16 + waves_per_block - 1) / waves_per_block;
        gc_gemm2_wmma<<<blocks, T, 0, stream>>>(agg1, W2, norm_d, b1, h2);
    }
    {
        long total = (long)N_EDGES * (OUT_F / 4);
        int blocks = (int)((total + T - 1) / T);
        gc_agg_kernel<<<blocks, T, 0, stream>>>(src, dst, h2, agg2, OUT_F);
    }
    gc_finalize_kernel<<<((long)N_NODES * OUT_F + T - 1) / T, T, 0, stream>>>(
        agg2, norm_d, b2, out);
}